// MyRNN_807453852061
// MI455X (gfx1250) — compile-verified
//
#include <hip/hip_runtime.h>
#include <cstdint>

typedef float v2f __attribute__((ext_vector_type(2)));
typedef float v8f __attribute__((ext_vector_type(8)));
typedef unsigned v4u __attribute__((ext_vector_type(4)));
typedef int v8i __attribute__((ext_vector_type(8)));
typedef int v4i __attribute__((ext_vector_type(4)));
typedef __attribute__((address_space(3))) float lds_float;

constexpr int Bn = 64, Tn = 512, DIN = 512, DH = 1024, DOUT = 512;
constexpr int LDSROW = 1026;   // 1024 dwords + 2-dword pad (bank-conflict fix)

// One fp32 WMMA: C(16x16) += A(16x4) * B(4x16).
// Fragment layout (ISA 7.12.2, 32-bit): lane = half*16 + l16,
//   A frag = {A[M0+l16][k+2*half], A[M0+l16][k+2*half+1]}
//   B frag = {W[N0+l16][k+2*half], W[N0+l16][k+2*half+1]}   (B = W^T, W row-major [N,K])
//   C/D: VGPR r -> row M0 + 8*half + r, col N0 + l16
__device__ __forceinline__ v8f wmma_f32(v2f a, v2f b, v8f c) {
    return __builtin_amdgcn_wmma_f32_16x16x4_f32(false, a, false, b,
                                                 (short)0, c, false, false);
}

// ---------------------------------------------------------------------------
// Phase 1: xp[(t*B+b), h] = x[b,t,:] . W_hx[h,:] + b_h[h]
// M = 32768, N = 1024, K = 512.  Wave tile = 32x32 (2x2 register blocking).
// ---------------------------------------------------------------------------
__global__ void __launch_bounds__(256)
xp_gemm_kernel(const float* __restrict__ x, const float* __restrict__ W_hx,
               const float* __restrict__ b_h, float* __restrict__ xp) {
    const int lane = threadIdx.x & 31;
    const int wave = threadIdx.x >> 5;
    const int mW = blockIdx.x >> 2;                  // 0..1023
    const int nW = ((blockIdx.x & 3) << 3) + wave;   // 0..31
    const int M0 = mW * 32, N0 = nW * 32;
    const int half = lane >> 4, l16 = lane & 15, kk0 = half * 2;

    const int m0 = M0 + l16, m1 = m0 + 16;           // rows in (t*B + b) order
    const float* a0p = x + ((size_t)(m0 & 63) * Tn + (m0 >> 6)) * DIN + kk0;
    const float* a1p = x + ((size_t)(m1 & 63) * Tn + (m1 >> 6)) * DIN + kk0;
    const float* w0p = W_hx + (size_t)(N0 + l16) * DIN + kk0;
    const float* w1p = W_hx + (size_t)(N0 + 16 + l16) * DIN + kk0;

    v8f c00 = {}, c01 = {}, c10 = {}, c11 = {};
#pragma unroll 4
    for (int k = 0; k < DIN; k += 4) {
        v2f a0 = *(const v2f*)(a0p + k), a1 = *(const v2f*)(a1p + k);
        v2f w0 = *(const v2f*)(w0p + k), w1 = *(const v2f*)(w1p + k);
        c00 = wmma_f32(a0, w0, c00);
        c01 = wmma_f32(a0, w1, c01);
        c10 = wmma_f32(a1, w0, c10);
        c11 = wmma_f32(a1, w1, c11);
    }

    const float bias0 = b_h[N0 + l16], bias1 = b_h[N0 + 16 + l16];
    float* s = xp + (size_t)(M0 + 8 * half) * DH + (N0 + l16);
#pragma unroll
    for (int r = 0; r < 8; ++r) {
        s[(size_t)r * DH]             = c00[r] + bias0;
        s[(size_t)r * DH + 16]        = c01[r] + bias1;
        s[(size_t)(r + 16) * DH]      = c10[r] + bias0;
        s[(size_t)(r + 16) * DH + 16] = c11[r] + bias1;
    }
}

// ---------------------------------------------------------------------------
// Phase 2: persistent scan kernel (32 blocks, one launch for all 512 steps).
// Each block owns a fixed 32-row slice of W_hh (128 KB); it is staged into
// LDS ONCE via the Tensor Data Mover (iterate mode: 32 rows, padded LDS row
// stride of 1026 dwords to dodge bank conflicts), then every step's
// B-fragments come from ds_load instead of 512x re-reading L2.
// Steps are separated by a device-scope atomic grid barrier.
// ---------------------------------------------------------------------------
__global__ void __launch_bounds__(256)
rnn_scan_kernel(const float* __restrict__ W_hh, float* __restrict__ xp,
                unsigned* __restrict__ counter) {
    __shared__ float wlds[32 * LDSROW];              // ~128.3 KB of 320 KB

    const int lane = threadIdx.x & 31;
    const int wave = threadIdx.x >> 5;
    const int g = blockIdx.x * 8 + wave;             // 0..255
    const int mTile = g & 3;                         // wave & 3
    const int nTile = g >> 2;                        // 2*blockIdx + (wave>=4)
    const int M0 = mTile * 16, N0 = nTile * 16;
    const int half = lane >> 4, l16 = lane & 15, kk0 = half * 2;
    const unsigned nBlk = gridDim.x;

    // ---- one-time TDM stage: W_hh rows [32*blockIdx, 32*blockIdx+32) ----
    if (wave == 0) {
        const unsigned long long ga =
            (unsigned long long)(const void*)(W_hh + (size_t)blockIdx.x * 32 * DH);
        const unsigned ldsOff = (unsigned)(uintptr_t)(lds_float*)wlds;

        v4u g0;
        g0[0] = 1u;                                  // count=1 (valid user D#)
        g0[1] = ldsOff;                              // lds_addr (bytes)
        g0[2] = (unsigned)ga;                        // global_addr[31:0]
        g0[3] = (unsigned)(ga >> 32) | (2u << 30);   // global_addr[56:32]|type=2

        v8i g1;
        g1[0] = (2 << 16) | (1 << 19);   // data_size=4B, iterate_enable=1
        g1[1] = (int)(1024u << 16);      // tensor_dim0 = 1024 (bits 79:48 lo)
        g1[2] = (int)(1u << 16);         // tensor_dim1 = 1    (bits 111:80 lo)
        g1[3] = (int)(1024u << 16);      // tile_dim0  = 1024  (bits 127:112)
        g1[4] = 1;                       // tile_dim1 = 1, tile_dim2 = 0
        g1[5] = 1024;                    // tensor_dim0_stride lo
        g1[6] = 0;                       // stride0 hi | stride1 lo
        g1[7] = 0;                       // stride1 hi

        v4i g2;
        g2[0] = 0;                       // tensor_dim2 (unused)
        g2[1] = LDSROW;                  // lds_addr_increment   (iterate mode)
        g2[2] = 1024;                    // global_addr_increment lo
        g2[3] = (int)(31u << 16);        // iterate_count = 31 -> 32 rows
        v4i g3 = {0, 0, 0, 0};
        v8i g4 = {0, 0, 0, 0, 0, 0, 0, 0};   // extra operand on this toolchain

        __builtin_amdgcn_tensor_load_to_lds(g0, g1, g2, g3, g4, 0);
        __builtin_amdgcn_s_wait_tensorcnt(0);
    }
    __syncthreads();                     // publish LDS W-slice to all waves

    // local W row for this lane's B fragment: (nTile - 2*blockIdx)*16 + l16
    const int lr = ((wave & 4) << 2) + l16;          // 0..31
    const float* wl = &wlds[lr * LDSROW + kk0];

    for (int t = 0; t < Tn; ++t) {
        float* slot = xp + (size_t)t * Bn * DH;
        float* cBase = slot + (size_t)(M0 + 8 * half) * DH + (N0 + l16);

        v8f c0, c1 = {};
#pragma unroll
        for (int r = 0; r < 8; ++r) c0[r] = cBase[(size_t)r * DH];

        if (t > 0) {  // uniform branch: EXEC stays all-ones for WMMA
            const float* aLane =
                slot - (size_t)Bn * DH + (size_t)(M0 + l16) * DH + kk0;
#pragma unroll 4
            for (int k = 0; k < DH; k += 8) {
                v2f a0 = *(const v2f*)(aLane + k);
                v2f w0 = *(const v2f*)(wl + k);
                v2f a1 = *(const v2f*)(aLane + k + 4);
                v2f w1 = *(const v2f*)(wl + k + 4);
                c0 = wmma_f32(a0, w0, c0);
                c1 = wmma_f32(a1, w1, c1);
            }
        }

#pragma unroll
        for (int r = 0; r < 8; ++r)
            cBase[(size_t)r * DH] = fmaxf(c0[r] + c1[r], 0.0f);

        // ---- device-scope grid barrier between timesteps ----
        __threadfence();                 // release our stores to agent scope
        __syncthreads();
        if (threadIdx.x == 0) {
            __hip_atomic_fetch_add(counter, 1u, __ATOMIC_RELAXED,
                                   __HIP_MEMORY_SCOPE_AGENT);
            const unsigned target = nBlk * (unsigned)(t + 1);
            while (__hip_atomic_load(counter, __ATOMIC_RELAXED,
                                     __HIP_MEMORY_SCOPE_AGENT) < target)
                __builtin_amdgcn_s_sleep(1);
        }
        __syncthreads();
        __threadfence();                 // acquire: invalidate stale lines
    }
}

// ---------------------------------------------------------------------------
// Phase 3: y[b,t,o] = hAll[(t*B+b), :] . W_yh[o, :] + b_y[o]
// M = 32768, N = 512, K = 1024.  Wave tile = 32x32 (2x2 blocking).
// ---------------------------------------------------------------------------
__global__ void __launch_bounds__(256)
y_gemm_kernel(const float* __restrict__ hAll, const float* __restrict__ W_yh,
              const float* __restrict__ b_y, float* __restrict__ out) {
    const int lane = threadIdx.x & 31;
    const int wave = threadIdx.x >> 5;
    const int mW = blockIdx.x >> 1;                  // 0..1023
    const int nW = ((blockIdx.x & 1) << 3) + wave;   // 0..15
    const int M0 = mW * 32, N0 = nW * 32;
    const int half = lane >> 4, l16 = lane & 15, kk0 = half * 2;

    const float* a0p = hAll + (size_t)(M0 + l16) * DH + kk0;
    const float* a1p = hAll + (size_t)(M0 + 16 + l16) * DH + kk0;
    const float* w0p = W_yh + (size_t)(N0 + l16) * DH + kk0;
    const float* w1p = W_yh + (size_t)(N0 + 16 + l16) * DH + kk0;

    v8f c00 = {}, c01 = {}, c10 = {}, c11 = {};
#pragma unroll 4
    for (int k = 0; k < DH; k += 4) {
        v2f a0 = *(const v2f*)(a0p + k), a1 = *(const v2f*)(a1p + k);
        v2f w0 = *(const v2f*)(w0p + k), w1 = *(const v2f*)(w1p + k);
        c00 = wmma_f32(a0, w0, c00);
        c01 = wmma_f32(a0, w1, c01);
        c10 = wmma_f32(a1, w0, c10);
        c11 = wmma_f32(a1, w1, c11);
    }

    const float bias0 = b_y[N0 + l16], bias1 = b_y[N0 + 16 + l16];
    const int col0 = N0 + l16, col1 = col0 + 16;
#pragma unroll
    for (int r = 0; r < 8; ++r) {
        const int mA = M0 + 8 * half + r;            // t*B + b
        const int mB = mA + 16;
        const size_t oA = ((size_t)(mA & 63) * Tn + (mA >> 6)) * DOUT;
        const size_t oB = ((size_t)(mB & 63) * Tn + (mB >> 6)) * DOUT;
        out[oA + col0] = c00[r] + bias0;
        out[oA + col1] = c01[r] + bias1;
        out[oB + col0] = c10[r] + bias0;
        out[oB + col1] = c11[r] + bias1;
    }
}

// ---------------------------------------------------------------------------
__global__ void init_counter_kernel(unsigned* __restrict__ c) { *c = 0u; }

extern "C" void kernel_launch(void* const* d_in, const int* in_sizes, int n_in,
                              void* d_out, int out_size, void* d_ws, size_t ws_size,
                              hipStream_t stream) {
    const float* x    = (const float*)d_in[0];
    const float* W_hx = (const float*)d_in[1];
    const float* W_hh = (const float*)d_in[2];
    const float* b_h  = (const float*)d_in[3];
    const float* W_yh = (const float*)d_in[4];
    const float* b_y  = (const float*)d_in[5];
    float* out = (float*)d_out;

    // Workspace: xp / h buffer [T][B][DH] fp32 = 128 MB, then barrier counter.
    float* xp = (float*)d_ws;
    unsigned* counter =
        (unsigned*)((char*)d_ws + (size_t)Tn * Bn * DH * sizeof(float));

    init_counter_kernel<<<1, 1, 0, stream>>>(counter);

    // Phase 1: all input projections (one big WMMA GEMM).
    xp_gemm_kernel<<<dim3(4096), 256, 0, stream>>>(x, W_hx, b_h, xp);

    // Phase 2: entire 512-step recurrence in ONE persistent kernel,
    // W_hh slice resident in LDS via the Tensor Data Mover.
    rnn_scan_kernel<<<dim3(32), 256, 0, stream>>>(W_hh, xp, counter);

    // Phase 3: output projection for all (t, b) rows at once.
    y_gemm_kernel<<<dim3(2048), 256, 0, stream>>>(xp, W_yh, b_y, out);
}